// MHack_3917010174103
// MI455X (gfx1250) — compile-verified
//
#include <hip/hip_runtime.h>
#include <cstdint>

// Problem constants from the reference: B=32, N=2048, ROWS=COLS=2.
#define NN            2048
#define NB            32
#define TILE          64
#define TSTRIDE       65            // padded LDS row stride (floats): conflict-free transposed reads
#define TILES_PER_DIM (NN / TILE)   // 32

typedef __attribute__((ext_vector_type(2))) float        v2f;
typedef __attribute__((ext_vector_type(8))) float        v8f;
typedef __attribute__((ext_vector_type(4))) unsigned int v4u;
typedef __attribute__((ext_vector_type(8))) unsigned int v8u;

// ---------------------------------------------------------------------------
// TDM: DMA one TILE x TILE f32 tile of x (global row stride = NN floats) into
// LDS, padding +1 DWORD after every 64-DWORD row => LDS row stride 65 floats.
// D# layout per CDNA5 ISA. Issued by one wave
// (EXEC ignored by tensor ops); tracked with TENSORcnt.
// ---------------------------------------------------------------------------
__device__ __forceinline__ void tdm_load_tile(const float* gp, unsigned lds_byte_addr)
{
    unsigned long long ga = (unsigned long long)(size_t)gp;
    v4u g0;
    g0[0] = 1u;                                                  // count=1, user, no gather
    g0[1] = lds_byte_addr;                                       // lds_addr (bytes)
    g0[2] = (unsigned)(ga & 0xFFFFFFFFull);                      // global_addr[31:0]
    g0[3] = (unsigned)((ga >> 32) & 0x1FFFFFFull) | (2u << 30);  // global_addr[56:32] | type=2
    v8u g1;
    g1[0] = (2u << 16)        // data_size = 4 bytes
          | (1u << 20)        // pad_enable
          | (5u << 22)        // pad_interval: 64 DWORDs (one tile row)
          | (0u << 25);       // pad_amount: 1 DWORD
    g1[1] = ((unsigned)NN)   << 16;   // tensor_dim0[15:0]   (bits 63:48)
    g1[2] = ((unsigned)NN)   << 16;   // tensor_dim0 hi = 0 | tensor_dim1[15:0]
    g1[3] = ((unsigned)TILE) << 16;   // tensor_dim1 hi = 0 | tile_dim0 = 64
    g1[4] = (unsigned)TILE;           // tile_dim1 = 64 | tile_dim2 = 0
    g1[5] = (unsigned)NN;             // tensor_dim0_stride[31:0] = 2048
    g1[6] = 0u;                       // stride hi | tensor_dim1_stride lo (unused, 2D)
    g1[7] = 0u;
    v4u g2 = {0u, 0u, 0u, 0u};        // dims 2..4 unused
    v4u g3 = {0u, 0u, 0u, 0u};
    asm volatile("tensor_load_to_lds %0, %1, %2, %3"
                 :
                 : "s"(g0), "s"(g1), "s"(g2), "s"(g3)
                 : "memory");
}

__device__ __forceinline__ void wait_tensorcnt0()
{
#if __has_builtin(__builtin_amdgcn_s_wait_tensorcnt)
    __builtin_amdgcn_s_wait_tensorcnt(0);
#else
    asm volatile("s_wait_tensorcnt 0x0" ::: "memory");
#endif
}

// ---------------------------------------------------------------------------
// Main kernel: block (i-tile = blockIdx.x, batch = blockIdx.y), 256 threads.
// TDM double-buffers x tiles (transposed consumption via padded LDS); M10/M11
// tiles come straight from L2 with coalesced b128 loads. Per-thread FMA
// accumulation, then a v_wmma_f32_16x16x4_f32-based wave reduction.
// ---------------------------------------------------------------------------
__global__ __launch_bounds__(256) void mhack_trace_kernel(
    const float* __restrict__ x, const float* __restrict__ Ms, float* __restrict__ partial)
{
    __shared__ float xt[2][TILE * TSTRIDE];   // 2 x 16.6 KB double buffer
    __shared__ float wred[8][2];

    const int  t   = threadIdx.x;
    const int  b   = blockIdx.y;
    const int  i0  = blockIdx.x * TILE;
    const int  jj4 = t & 15;                  // float4 index along j within tile
    const int  iiB = t >> 4;                  // base i within tile (0..15)
    const bool w0  = (t >> 5) == 0;           // wave 0 drives the TDM

    const float* xb  = x  + (size_t)b * NN * NN;
    const float* M10 = Ms + (size_t)2 * NN * NN;   // Ms[1,0]
    const float* M11 = Ms + (size_t)3 * NN * NN;   // Ms[1,1]

    float acc10 = 0.f, acc11 = 0.f;

    if (w0) tdm_load_tile(xb + i0, (unsigned)(size_t)&xt[0][0]);

    int buf = 0;
    for (int jt = 0; jt < TILES_PER_DIM; ++jt) {
        if (w0) wait_tensorcnt0();            // tile jt landed in xt[buf]
        __syncthreads();                      // publish to all waves; prior compute done
        if (w0 && (jt + 1) < TILES_PER_DIM)   // prefetch tile jt+1 into other buffer
            tdm_load_tile(xb + (size_t)(jt + 1) * TILE * NN + i0,
                          (unsigned)(size_t)&xt[buf ^ 1][0]);

        const int    j0   = jt * TILE;
        const float* tile = &xt[buf][0];      // tile[j_local*65 + i_local] = x[b, j0+jl, i0+il]
#pragma unroll
        for (int p = 0; p < 4; ++p) {
            const int    ii   = iiB + (p << 4);
            const size_t mrow = (size_t)(i0 + ii) * NN + (size_t)(j0 + (jj4 << 2));
            const float4 m10  = *(const float4*)(M10 + mrow);   // coalesced, L2-resident
            const float4 m11  = *(const float4*)(M11 + mrow);
            const int    lb   = (jj4 << 2) * TSTRIDE + ii;      // bank-conflict-free column read
            float xv;
            xv = tile[lb];               acc10 = fmaf(m10.x, xv, acc10); acc11 = fmaf(m11.x, xv * xv, acc11);
            xv = tile[lb +   TSTRIDE];   acc10 = fmaf(m10.y, xv, acc10); acc11 = fmaf(m11.y, xv * xv, acc11);
            xv = tile[lb + 2*TSTRIDE];   acc10 = fmaf(m10.z, xv, acc10); acc11 = fmaf(m11.z, xv * xv, acc11);
            xv = tile[lb + 3*TSTRIDE];   acc10 = fmaf(m10.w, xv, acc10); acc11 = fmaf(m11.w, xv * xv, acc11);
        }
        buf ^= 1;
    }

    // ---- wave reduction: one WMMA pair-sums lanes n / n+16 and separates the
    // two accumulators into D rows 0 and 1 (A = selector, B = {acc10, acc11}).
    // A 16x4 layout: lanes0-15 {a0=K0,a1=K1}, lanes16-31 {a0=K2,a1=K3}.
    // B  4x16 layout: b0 = rows K0|K2, b1 = rows K1|K3, column = lane%16.
#if __has_builtin(__builtin_amdgcn_wmma_f32_16x16x4_f32)
    {
        const int lane = t & 31;
        v2f a;  a[0] = (lane == 0 || lane == 16) ? 1.f : 0.f;   // A[0][k] = k even
                a[1] = (lane == 1 || lane == 17) ? 1.f : 0.f;   // A[1][k] = k odd
        v2f bm; bm[0] = acc10; bm[1] = acc11;
        v8f c = {0.f, 0.f, 0.f, 0.f, 0.f, 0.f, 0.f, 0.f};
        v8f d = __builtin_amdgcn_wmma_f32_16x16x4_f32(
            false, a, false, bm, (short)0, c, false, false);
        acc10 = d[0];   // lanes 0-15: D[0][n] = acc10_n + acc10_{n+16}; lanes 16-31: 0
        acc11 = d[1];   // lanes 0-15: D[1][n] = acc11_n + acc11_{n+16}; lanes 16-31: 0
    }
#endif
    float r10 = acc10, r11 = acc11;
#pragma unroll
    for (int off = 16; off >= 1; off >>= 1) {   // butterfly finishes the wave sum
        r10 += __shfl_xor(r10, off, 32);
        r11 += __shfl_xor(r11, off, 32);
    }
    if ((t & 31) == 0) { wred[t >> 5][0] = r10; wred[t >> 5][1] = r11; }
    __syncthreads();
    if (t == 0) {
        float s10 = 0.f, s11 = 0.f;
#pragma unroll
        for (int w = 0; w < 8; ++w) { s10 += wred[w][0]; s11 += wred[w][1]; }
        const size_t idx = ((size_t)b * TILES_PER_DIM + blockIdx.x) * 2;
        partial[idx]     = s10;
        partial[idx + 1] = s11;
    }
}

// ---------------------------------------------------------------------------
// Finalize: fold block partials, apply coef and sigmoid.
// out[b] = sigmoid(coef[0] + coef[1] + coef[2]*tr10 + coef[3]*tr11)
// (coef flat (COLS,ROWS) broadcasts so coef_flat[r*2+c] multiplies tr[b,r,c];
//  r=0 terms are tr**0 == 1.)
// ---------------------------------------------------------------------------
__global__ void mhack_finalize_kernel(const float* __restrict__ partial,
                                      const float* __restrict__ coef,
                                      float* __restrict__ out)
{
    const int b = threadIdx.x;
    if (b < NB) {
        float s10 = 0.f, s11 = 0.f;
        for (int ti = 0; ti < TILES_PER_DIM; ++ti) {
            const size_t idx = ((size_t)b * TILES_PER_DIM + ti) * 2;
            s10 += partial[idx];
            s11 += partial[idx + 1];
        }
        const float z = coef[0] + coef[1] + coef[2] * s10 + coef[3] * s11;
        out[b] = 1.f / (1.f + __expf(-z));
    }
}

extern "C" void kernel_launch(void* const* d_in, const int* in_sizes, int n_in,
                              void* d_out, int out_size, void* d_ws, size_t ws_size,
                              hipStream_t stream)
{
    const float* x    = (const float*)d_in[0];   // (32, 2048, 2048) f32
    const float* Ms   = (const float*)d_in[1];   // (2, 2, 2048, 2048) f32
    const float* coef = (const float*)d_in[2];   // (2, 2) f32
    float*       out  = (float*)d_out;           // (32,) f32
    float*       part = (float*)d_ws;            // 32*32*2 floats = 8 KB scratch

    dim3 grid(TILES_PER_DIM, NB);                // (i-tile, batch) = 1024 blocks
    mhack_trace_kernel<<<grid, 256, 0, stream>>>(x, Ms, part);
    mhack_finalize_kernel<<<1, 32, 0, stream>>>(part, coef, out);
}